// CTCGreedyDecoder_28226525070317
// MI455X (gfx1250) — compile-verified
//
#include <hip/hip_runtime.h>

#define B_ 64
#define T_ 2048
#define V_ 512

typedef float v4f __attribute__((ext_vector_type(4)));
typedef int   v4i __attribute__((ext_vector_type(4)));

// ---------------------------------------------------------------------------
// Kernel 1: argmax over V=512 per (b,t). One wave32 per row (2KB contiguous).
// Non-temporal b128 loads (256MB streamed once, > 192MB L2 -> NT is correct).
// First-occurrence tie-break to match jnp.argmax.
// ---------------------------------------------------------------------------
__global__ __launch_bounds__(256) void argmax512_kernel(const float* __restrict__ probs,
                                                        int* __restrict__ tokens) {
    const int lane = threadIdx.x & 31;
    const int waveInBlk = threadIdx.x >> 5;
    const long long row = (long long)blockIdx.x * 8 + waveInBlk;  // 8 waves per block

    const v4f* __restrict__ p =
        reinterpret_cast<const v4f*>(probs + row * (long long)V_);

    float best = -3.402823466e+38f;
    int bidx = 0;
#pragma unroll
    for (int j = 0; j < 4; ++j) {
        const int vi = j * 32 + lane;                  // float4 index within row
        v4f v = __builtin_nontemporal_load(&p[vi]);    // global_load_b128 th:NT
        const int i0 = vi * 4;
        if (v.x > best) { best = v.x; bidx = i0;     }
        if (v.y > best) { best = v.y; bidx = i0 + 1; }
        if (v.z > best) { best = v.z; bidx = i0 + 2; }
        if (v.w > best) { best = v.w; bidx = i0 + 3; }
    }
    // wave32 butterfly reduction; lexicographic max of (value, -index)
#pragma unroll
    for (int off = 16; off > 0; off >>= 1) {
        float ov = __shfl_xor(best, off, 32);
        int   oi = __shfl_xor(bidx, off, 32);
        if (ov > best || (ov == best && oi < bidx)) { best = ov; bidx = oi; }
    }
    if (lane == 0) tokens[row] = bidx;
}

// ---------------------------------------------------------------------------
// Kernel 2: per-row run-dedup + blank-drop + compaction. One block per row.
// Token row staged in LDS via CDNA5 async global->LDS loads when available.
// ---------------------------------------------------------------------------
__global__ __launch_bounds__(256) void ctc_compact_kernel(const int* __restrict__ tokens,
                                                          const int* __restrict__ lengths,
                                                          float* __restrict__ out) {
    const int b   = blockIdx.x;
    const int tid = threadIdx.x;

    __shared__ int s_tok[T_];    // raw token row
    __shared__ int s_cmp[T_];    // compacted row (zero-padded)
    __shared__ int s_wave[8];
    __shared__ int s_total;

    const int* __restrict__ row = tokens + (long long)b * T_;

#if __has_builtin(__builtin_amdgcn_global_load_async_to_lds_b128)
    {
        typedef __attribute__((address_space(1))) v4i* gp_t;  // global (prints __device__)
        typedef __attribute__((address_space(3))) v4i* lp_t;  // LDS    (prints __shared__)
        // 8 ints per thread = 2 x b128 async transfers, 16B aligned both sides
        __builtin_amdgcn_global_load_async_to_lds_b128(
            (gp_t)(row + tid * 8),     (lp_t)(&s_tok[tid * 8]),     0, 0);
        __builtin_amdgcn_global_load_async_to_lds_b128(
            (gp_t)(row + tid * 8 + 4), (lp_t)(&s_tok[tid * 8 + 4]), 0, 0);
#if __has_builtin(__builtin_amdgcn_s_wait_asynccnt)
        __builtin_amdgcn_s_wait_asynccnt(0);
#else
        asm volatile("s_wait_asynccnt 0" ::: "memory");
#endif
    }
#else
    {
        const int4* r4 = reinterpret_cast<const int4*>(row);
        int4 a = r4[tid * 2];
        int4 c = r4[tid * 2 + 1];
        reinterpret_cast<int4*>(s_tok)[tid * 2]     = a;
        reinterpret_cast<int4*>(s_tok)[tid * 2 + 1] = c;
    }
#endif
    // zero the compact buffer
    reinterpret_cast<int4*>(s_cmp)[tid * 2]     = make_int4(0, 0, 0, 0);
    reinterpret_cast<int4*>(s_cmp)[tid * 2 + 1] = make_int4(0, 0, 0, 0);
    const int len = lengths[b];
    __syncthreads();

    // keep[i] = (i < len) && (tok[i] != tok[i-1], sentinel -1 at i==0) && tok[i] != 0
    const int base = tid * 8;
    int prev = (base == 0) ? -1 : s_tok[base - 1];
    int toks[8];
    int keepmask = 0, cnt = 0;
#pragma unroll
    for (int k = 0; k < 8; ++k) {
        const int tok = s_tok[base + k];
        toks[k] = tok;
        const bool kp = ((base + k) < len) && (tok != prev) && (tok != 0);
        keepmask |= ((int)kp) << k;
        cnt += (int)kp;
        prev = tok;
    }

    // wave32 inclusive scan of per-thread counts
    const int lane = tid & 31, wv = tid >> 5;
    int incl = cnt;
#pragma unroll
    for (int off = 1; off < 32; off <<= 1) {
        int n = __shfl_up(incl, off, 32);
        if (lane >= off) incl += n;
    }
    if (lane == 31) s_wave[wv] = incl;
    __syncthreads();
    if (tid == 0) {
        int acc = 0;
#pragma unroll
        for (int w = 0; w < 8; ++w) { int v = s_wave[w]; s_wave[w] = acc; acc += v; }
        s_total = acc;
    }
    __syncthreads();

    // scatter kept tokens into LDS compact buffer
    int dst = s_wave[wv] + (incl - cnt);
#pragma unroll
    for (int k = 0; k < 8; ++k) {
        if ((keepmask >> k) & 1) s_cmp[dst++] = toks[k];
    }
    __syncthreads();

    // stream finished row to global as floats (contiguous per thread)
    float* __restrict__ orow = out + (long long)b * T_;
    float4 o0 = make_float4((float)s_cmp[base + 0], (float)s_cmp[base + 1],
                            (float)s_cmp[base + 2], (float)s_cmp[base + 3]);
    float4 o1 = make_float4((float)s_cmp[base + 4], (float)s_cmp[base + 5],
                            (float)s_cmp[base + 6], (float)s_cmp[base + 7]);
    reinterpret_cast<float4*>(orow)[tid * 2]     = o0;
    reinterpret_cast<float4*>(orow)[tid * 2 + 1] = o1;

    if (tid == 0) out[(long long)B_ * T_ + b] = (float)s_total;
}

// ---------------------------------------------------------------------------
extern "C" void kernel_launch(void* const* d_in, const int* in_sizes, int n_in,
                              void* d_out, int out_size, void* d_ws, size_t ws_size,
                              hipStream_t stream) {
    const float* probs   = (const float*)d_in[0];   // [B, T, V] f32
    const int*   lengths = (const int*)d_in[1];     // [B] int
    float*       out     = (float*)d_out;           // [B*T] out ++ [B] out_lengths
    int*         tokens  = (int*)d_ws;              // [B*T] int scratch (512 KB)

    // Kernel 1: one wave per (b,t) row; 8 waves per 256-thread block.
    const int nrows  = B_ * T_;            // 131072
    const int blocks = nrows / 8;          // 16384
    argmax512_kernel<<<blocks, 256, 0, stream>>>(probs, tokens);

    // Kernel 2: one block per batch row.
    ctc_compact_kernel<<<B_, 256, 0, stream>>>(tokens, lengths, out);
}